// PatternMatchingInference_8907762172208
// MI455X (gfx1250) — compile-verified
//
#include <hip/hip_runtime.h>
#include <hip/hip_bf16.h>
#include <stdint.h>
#include <math.h>

#define S_  69
#define C_  4
#define CS  (C_ * S_)          // 276 states
#define B_  512
#define STEPS 68
#define NT  288                 // 9 wave32s, >= 276 states
#define NW  (NT / 32)           // 9 waves
#define CSTRIDE (C_ * S_ * S_)  // 19044: cprev stride in global Q
#define QDW (C_ * C_ * S_ * S_) // 76176 dwords of Q
#define QV4 (QDW / 4)           // 19044 16-byte chunks
#define STAGE_V4 512            // 8KB staging buffer (float4 granules)

typedef __attribute__((ext_vector_type(2))) float v2f;
typedef __attribute__((ext_vector_type(8))) float v8f;

__device__ __forceinline__ unsigned lds_off(const void* p) {
  // flat pointers into the LDS aperture carry the LDS byte offset in [31:0]
  return (unsigned)(uintptr_t)p;
}

// ---------------------------------------------------------------------------
// Forward scan: one block per batch element.
//  - P[b, cn, :, sn] in VGPRs (step-invariant, read from L2 once)
//  - T*Q in LDS, TRANSPOSED to Qt[cn][sp][sn][cp]: one ds_load_b128 per sp;
//    filled via chunked CDNA5 async global->LDS DMA (ASYNCcnt-tracked)
//  - la kept as la4[sp][cp] float4 (broadcast b128 reads)
//  - logsumexp Sum(exp) on the matrix pipe (v_wmma_f32_16x16x4_f32),
//    executed redundantly by ALL waves so no lse broadcast barrier is needed
// Writes raw normalized alphas time-reversed into out_alpha, norms to ws.
// ---------------------------------------------------------------------------
__global__ __launch_bounds__(NT)
void fwd_kernel(const float* __restrict__ P, const float* __restrict__ Q,
                const float* __restrict__ pi, const int* __restrict__ Tp,
                float* __restrict__ out_alpha,   // [B][69][276]
                float* __restrict__ norms_ws)    // [69][B]
{
  __shared__ alignas(16) float Qt[QDW];            // 304,704 B transposed T*Q
  __shared__ alignas(16) float stage[STAGE_V4*4];  // 8 KB async staging
  __shared__ alignas(16) float la4[S_ * C_];       // previous alpha, [sp][cp]
  __shared__ alignas(16) float sume[320];          // exp values, 5 WMMA tiles
  __shared__ float redw[NW];                       // per-wave max partials

  const int b = blockIdx.x;
  const int t = threadIdx.x;
  const int lane = t & 31;
  const int wid  = t >> 5;
  const bool act = t < CS;
  const int tc = act ? t : (CS - 1);
  const int cn = tc / S_;
  const int sn = tc % S_;
  const float Tf = (float)Tp[0];

  // ---- chunked CDNA5 async DMA of Q into stage, scatter-transpose into Qt --
  {
    const unsigned stage_base = lds_off(stage);
    for (int v4base = 0; v4base < QV4; v4base += STAGE_V4) {
      const int n_v4 = min(STAGE_V4, QV4 - v4base);
      for (int i = t; i < n_v4; i += NT) {
        unsigned goff = (unsigned)(v4base + i) * 16u;  // byte offset into Q
        unsigned l    = stage_base + (unsigned)i * 16u;
        asm volatile("global_load_async_to_lds_b128 %0, %1, %2"
                     :: "v"(l), "v"(goff), "s"(Q) : "memory");
      }
      asm volatile("s_wait_asynccnt 0x0" ::: "memory");
      __syncthreads();
      const int n_dw = n_v4 * 4;
      for (int i = t; i < n_dw; i += NT) {
        const int g  = v4base * 4 + i;        // global dword index into Q
        const int cp = g / CSTRIDE;           // source: [cp][cn][sp][sn]
        const int r  = g % CSTRIDE;           // = (cn*S+sp)*S+sn
        Qt[r * 4 + cp] = Tf * stage[i];       // dest: [cn][sp][sn][cp]
      }
      __syncthreads();                        // all reads of stage done
    }
  }
  if (t < 32) sume[288 + t] = 0.0f;           // pad WMMA reduction tail once

  // ---- step-invariant P[b, cn, :, sn] into registers ----
  float Pr[S_];
  {
    const float* Pp = P + (((size_t)b * C_ + cn) * S_) * (size_t)S_ + sn;
    #pragma unroll
    for (int sp = 0; sp < S_; ++sp) Pr[sp] = Pp[(size_t)sp * S_];
  }

  // ---- initial state: a0 = -inf except column 0 = pi; n0 = logsumexp(pi) ----
  const float p0 = pi[0], p1 = pi[1], p2 = pi[2], p3 = pi[3];
  const float pm = fmaxf(fmaxf(p0, p1), fmaxf(p2, p3));
  const float n0 = pm + logf(expf(p0 - pm) + expf(p1 - pm) +
                             expf(p2 - pm) + expf(p3 - pm));
  float nrun = n0;
  {
    const float a0 = (act && sn == 0) ? (pi[cn] - n0) : -INFINITY;
    if (act) {
      la4[sn * C_ + cn] = a0;
      out_alpha[((size_t)b * (STEPS + 1) + STEPS) * CS + t] = a0;  // reversed
    }
    if (t == 0) norms_ws[0 * B_ + b] = n0;
  }
  __syncthreads();

  const float* qcol = &Qt[(cn * S_ * S_ + sn) * 4];   // + sp*S_*4 per step
  const int m16 = t & 15;
  const int hi  = (t >> 4) & 1;     // bit4 of lane: lanes 16-31 carry K=2,3

  #pragma unroll 1
  for (int j = 1; j <= STEPS; ++j) {
    // max-plus contraction over (cprev, sprev): b128 Q + broadcast b128 la
    float v = -INFINITY;
    #pragma unroll
    for (int sp = 0; sp < S_; ++sp) {
      const float4 q = *reinterpret_cast<const float4*>(qcol + (size_t)sp * (S_ * 4));
      const float4 l = *reinterpret_cast<const float4*>(&la4[sp * 4]);
      const float g = fmaxf(fmaxf(q.x + l.x, q.y + l.y),
                            fmaxf(q.z + l.z, q.w + l.w));
      v = fmaxf(v, Pr[sp] + g);
    }

    // ---- block max: per-wave shfl tree, one float per wave through LDS ----
    float m = act ? v : -INFINITY;
    #pragma unroll
    for (int off = 16; off > 0; off >>= 1)
      m = fmaxf(m, __shfl_xor(m, off, 32));
    if (lane == 0) redw[wid] = m;
    __syncthreads();                               // barrier #1
    float M = redw[0];
    #pragma unroll
    for (int w = 1; w < NW; ++w) M = fmaxf(M, redw[w]);

    // ---- Sum(exp) on the matrix pipe, redundantly on every wave ----
    sume[t] = act ? expf(v - M) : 0.0f;
    __syncthreads();                               // barrier #2
    v8f acc = {0.f, 0.f, 0.f, 0.f, 0.f, 0.f, 0.f, 0.f};
    v2f bones; bones.x = 1.0f; bones.y = 1.0f;
    #pragma unroll
    for (int tile = 0; tile < 5; ++tile) {
      const float* tb = &sume[tile * 64 + hi * 32];
      v2f a; a.x = tb[m16]; a.y = tb[16 + m16];
      acc = __builtin_amdgcn_wmma_f32_16x16x4_f32(
                false, a, false, bones, (short)0, acc, false, false);
    }
    const float psum = acc[0] + acc[1] + acc[2] + acc[3] +
                       acc[4] + acc[5] + acc[6] + acc[7];       // rows 0-7 / 8-15
    const float tot = __shfl(psum, 0, 32) + __shfl(psum, 16, 32); // all 16 rows
    const float lse = M + logf(tot);
    nrun += lse;
    const float anew = v - lse;

    if (act) {
      la4[sn * C_ + cn] = anew;
      out_alpha[((size_t)b * (STEPS + 1) + (STEPS - j)) * CS + t] = anew;
    }
    if (t == 0) norms_ws[j * B_ + b] = nrun;
    __syncthreads();                               // barrier #3
  }
}

// ---------------------------------------------------------------------------
// Fold norms[k] into the time-reversed alphas (float4-vectorized; CS % 4 == 0)
// ---------------------------------------------------------------------------
__global__ void addnorm_kernel(float* __restrict__ out_alpha,
                               const float* __restrict__ norms_ws)
{
  const size_t total4 = (size_t)B_ * (STEPS + 1) * (CS / 4);
  size_t idx = (size_t)blockIdx.x * blockDim.x + threadIdx.x;
  if (idx >= total4) return;
  const size_t per_b4 = (size_t)(STEPS + 1) * (CS / 4);
  const int b = (int)(idx / per_b4);
  const int k = (int)((idx % per_b4) / (CS / 4));
  const float n = norms_ws[k * B_ + b];
  float4* p = reinterpret_cast<float4*>(out_alpha) + idx;
  float4 x = *p;
  x.x += n; x.y += n; x.z += n; x.w += n;
  *p = x;
}

// ---------------------------------------------------------------------------
// Backtrack: one wave32 per batch element, 68 sequential argmax steps over
// 276 candidates, first-index tie-break to match jnp.argmax.
// ---------------------------------------------------------------------------
__global__ __launch_bounds__(32)
void bt_kernel(const float* __restrict__ P, const float* __restrict__ Q,
               const int* __restrict__ ls, const int* __restrict__ Tp,
               const float* __restrict__ out_alpha,
               float* __restrict__ out_pat)
{
  const int b = blockIdx.x;
  const int lane = threadIdx.x;
  const float Tf = (float)Tp[0];
  int c = 3;
  int tt = ls[b];
  if (lane == 0) {
    out_pat[(size_t)b * (STEPS + 1) * 2 + STEPS * 2 + 0] = 3.0f;
    out_pat[(size_t)b * (STEPS + 1) * 2 + STEPS * 2 + 1] = (float)tt;
  }
  for (int k = 0; k < STEPS; ++k) {
    const float* a = out_alpha + ((size_t)b * (STEPS + 1) + (k + 1)) * CS;
    float best = -INFINITY; int bidx = CS;
    for (int j = lane; j < CS; j += 32) {
      const int cp = j / S_, sp = j % S_;
      const float val = P[(((size_t)b  * C_ + c) * S_ + sp) * S_ + tt]
               + Tf *   Q[(((size_t)cp * C_ + c) * S_ + sp) * S_ + tt]
               + a[j];
      if (val > best) { best = val; bidx = j; }
    }
    #pragma unroll
    for (int off = 16; off > 0; off >>= 1) {
      const float ob = __shfl_xor(best, off, 32);
      const int   oi = __shfl_xor(bidx, off, 32);
      if (ob > best || (ob == best && oi < bidx)) { best = ob; bidx = oi; }
    }
    bidx = __shfl(bidx, 0, 32);
    c = bidx / S_; tt = bidx % S_;
    if (lane == 0) {
      out_pat[(size_t)b * (STEPS + 1) * 2 + (STEPS - 1 - k) * 2 + 0] = (float)c;
      out_pat[(size_t)b * (STEPS + 1) * 2 + (STEPS - 1 - k) * 2 + 1] = (float)tt;
    }
  }
}

// ---------------------------------------------------------------------------
extern "C" void kernel_launch(void* const* d_in, const int* in_sizes, int n_in,
                              void* d_out, int out_size, void* d_ws, size_t ws_size,
                              hipStream_t stream)
{
  const float* P  = (const float*)d_in[0];
  const float* Q  = (const float*)d_in[1];
  const float* pi = (const float*)d_in[2];
  const int*   ls = (const int*)  d_in[3];
  const int*   Tp = (const int*)  d_in[4];

  float* out       = (float*)d_out;
  float* out_pat   = out;                                   // B*69*2 (ints as f32)
  float* out_alpha = out + (size_t)B_ * (STEPS + 1) * 2;    // B*69*4*69
  float* norms_ws  = (float*)d_ws;                          // 69*B floats

  fwd_kernel<<<B_, NT, 0, stream>>>(P, Q, pi, Tp, out_alpha, norms_ws);

  const size_t total4 = (size_t)B_ * (STEPS + 1) * (CS / 4);
  const int thr = 256;
  addnorm_kernel<<<(int)((total4 + thr - 1) / thr), thr, 0, stream>>>(out_alpha, norms_ws);

  bt_kernel<<<B_, 32, 0, stream>>>(P, Q, ls, Tp, out_alpha, out_pat);
}